// DereverbWPE_1726576856353
// MI455X (gfx1250) — compile-verified
//
#include <hip/hip_runtime.h>
#include <math.h>

// ---------------------------------------------------------------------------
// WPE dereverberation for MI455X (gfx1250), fp32 WMMA 16x16x4 everywhere.
// ---------------------------------------------------------------------------

typedef float v2f __attribute__((ext_vector_type(2)));
typedef float v8f __attribute__((ext_vector_type(8)));

constexpr int NB    = 4;       // batch
constexpr int DCH   = 8;       // channels
constexpr int TLEN  = 32768;   // samples
constexpr int NFFT  = 512;
constexpr int HOP   = 128;
constexpr int NF    = 257;     // freq bins
constexpr int NTF   = 257;     // frames
constexpr int TFP   = 260;     // frames padded to mult of 4 (WMMA K)
constexpr int TAPS  = 10;
constexpr int DELAY = 3;
constexpr int KDIM  = 80;      // taps * D
constexpr int ITERS = 3;

// workspace layout (float offsets)
constexpr size_t OFF_WIN  = 0;                            // 512
constexpr size_t OFF_CRE  = 512;                          // 272*512 (f padded to 272)
constexpr size_t OFF_CIM  = OFF_CRE + 272 * 512;
constexpr size_t OFF_ICRE = OFF_CIM + 272 * 512;          // 512*260 (f padded to 260)
constexpr size_t OFF_ICIM = OFF_ICRE + 512 * 260;
constexpr size_t OFF_Y    = OFF_ICIM + 512 * 260;         // (B,F,D,Tf) float2
constexpr size_t YFLOATS  = (size_t)NB * NF * DCH * NTF * 2;
constexpr size_t OFF_X    = OFF_Y + YFLOATS;
constexpr size_t OFF_POW  = OFF_X + YFLOATS;              // (B,F,Tf)
constexpr size_t OFF_MAX  = OFF_POW + ((size_t)NB * NF * NTF + 7) / 8 * 8;
constexpr size_t OFF_WF   = OFF_Y;                        // frames buffer aliases dead Y

__device__ inline v8f wmma4(v2f a, v2f b, v8f c) {
  // V_WMMA_F32_16X16X4_F32 : D = A(16x4) * B(4x16) + C
  return __builtin_amdgcn_wmma_f32_16x16x4_f32(false, a, false, b, (short)0, c,
                                               false, false);
}

__device__ inline float winf(int n) {
  float a = 6.28318530718f * (float)n * (1.0f / 512.0f);
  return 0.42f - 0.5f * __cosf(a) + 0.08f * __cosf(2.0f * a);
}
__device__ inline float angf(int fn) {  // 2*pi*(fn mod 512)/512 : exact periodicity
  return 6.28318530718f * (float)(fn & 511) * (1.0f / 512.0f);
}

// ---------------------------------------------------------------------------
// Init: window, forward DFT matrix (cos/sin), inverse DFT matrix with window,
// 2/N weights folded in.
__global__ void k_init(float* __restrict__ win, float* __restrict__ Cre,
                       float* __restrict__ Cim, float* __restrict__ ICre,
                       float* __restrict__ ICim) {
  int gid = blockIdx.x * 256 + threadIdx.x;
  if (gid < NFFT) win[gid] = winf(gid);
  if (gid < 272 * NFFT) {
    int f = gid / NFFT, n = gid % NFFT;
    float c = 0.f, s = 0.f;
    if (f < NF) { float a = angf(f * n); c = __cosf(a); s = __sinf(a); }
    Cre[gid] = c; Cim[gid] = s;
  }
  if (gid < NFFT * TFP) {
    int n = gid / TFP, f = gid % TFP;
    float c = 0.f, s = 0.f;
    if (f < NF) {
      float wf = (f == 0 || f == NF - 1) ? 1.0f : 2.0f;
      float sc = winf(n) * wf * (1.0f / (float)NFFT);
      float a = angf(f * n);
      c = sc * __cosf(a); s = sc * __sinf(a);
    }
    ICre[gid] = c; ICim[gid] = s;
  }
}

// ---------------------------------------------------------------------------
// STFT as WMMA GEMM: Y(F x Tf) = C(F x 512) * Frames(512 x Tf) per (b,d).
// One 16x16 output tile per wave; 17x17 tiles; reflect padding per lane.
__global__ __launch_bounds__(256) void k_stft(const float* __restrict__ x,
                                              const float* __restrict__ win,
                                              const float* __restrict__ Cre,
                                              const float* __restrict__ Cim,
                                              float2* __restrict__ Yg) {
  const int tid = threadIdx.x, wave = tid >> 5, lane = tid & 31;
  const int gj = blockIdx.x * 8 + wave;  // 32 * 289 jobs total
  const int bd = gj / 289;
  const int tile = gj % 289;
  const int fm = (tile / 17) * 16;
  const int tn = (tile % 17) * 16;
  const int b = bd >> 3, d = bd & 7;
  const int mrow = lane & 15, kb2 = (lane < 16) ? 0 : 2;
  const float* xs = x + (size_t)bd * TLEN;

  v8f accRe = {0, 0, 0, 0, 0, 0, 0, 0};
  v8f accIm = {0, 0, 0, 0, 0, 0, 0, 0};
  const int t = tn + mrow;
  for (int s = 0; s < 128; ++s) {
    const int n0 = 4 * s + kb2;
    v2f Are = {Cre[(fm + mrow) * NFFT + n0], Cre[(fm + mrow) * NFFT + n0 + 1]};
    v2f Aim = {Cim[(fm + mrow) * NFFT + n0], Cim[(fm + mrow) * NFFT + n0 + 1]};
    int p0 = t * HOP + n0 - 256;
    int p1 = p0 + 1;
    if (p0 < 0) p0 = -p0;
    if (p0 >= TLEN) p0 = 2 * TLEN - 2 - p0;
    if (p1 < 0) p1 = -p1;
    if (p1 >= TLEN) p1 = 2 * TLEN - 2 - p1;
    v2f Bf = {xs[p0] * win[n0], xs[p1] * win[n0 + 1]};
    accRe = wmma4(Are, Bf, accRe);
    accIm = wmma4(Aim, Bf, accIm);
  }
  const int col = mrow, rb = (lane < 16) ? 0 : 8;
  const int tt = tn + col;
#pragma unroll
  for (int i = 0; i < 8; ++i) {
    int fr = fm + rb + i;
    if (fr < NF && tt < NTF)
      Yg[((size_t)(b * NF + fr) * DCH + d) * NTF + tt] =
          make_float2(accRe[i], -accIm[i]);  // exp(-i..): Im = -sin part
  }
}

// ---------------------------------------------------------------------------
// Power = mean_d |X|^2 ; per-batch max via bitwise atomicMax (exact, p>=0).
__global__ __launch_bounds__(256) void k_power(const float2* __restrict__ src,
                                               float* __restrict__ power,
                                               unsigned* __restrict__ maxb) {
  __shared__ float red[256];
  const int bf = blockIdx.x;
  const int b = bf / NF;
  const int tid = threadIdx.x;
  const size_t base = (size_t)bf * DCH * NTF;
  float lmax = 0.f;
  for (int t = tid; t < NTF; t += 256) {
    float acc = 0.f;
    for (int d = 0; d < DCH; ++d) {
      float2 v = src[base + (size_t)d * NTF + t];
      acc += v.x * v.x + v.y * v.y;
    }
    float p = acc * (1.0f / (float)DCH);
    power[(size_t)bf * NTF + t] = p;
    lmax = fmaxf(lmax, p);
  }
  red[tid] = lmax;
  __syncthreads();
  for (int s = 128; s > 0; s >>= 1) {
    if (tid < s) red[tid] = fmaxf(red[tid], red[tid + s]);
    __syncthreads();
  }
  if (tid == 0) atomicMax(maxb + b, __float_as_uint(red[0]));
}

__global__ void k_zeromax(unsigned* maxb) {
  if (threadIdx.x < NB) maxb[threadIdx.x] = 0u;
}

// ---------------------------------------------------------------------------
// Fused WPE per (b,f): build R(80x80)+P(80x8) complex with f32 WMMA in LDS,
// Gaussian solve in LDS, apply filter X = Y - G^H Yt. Yt is an implicit
// time-shifted view of Y (never materialized).
__global__ __launch_bounds__(256) void k_wpe(const float2* __restrict__ Yg,
                                             const float* __restrict__ power,
                                             const unsigned* __restrict__ maxb,
                                             float2* __restrict__ Xg) {
  __shared__ float2 sY[DCH][TFP];    // staged Y for this (b,f), zero padded
  __shared__ float sInvp[TFP];       // 1/max(power,eps), 0 in padding
  __shared__ float2 sR[KDIM][KDIM];  // correlation matrix
  __shared__ float2 sP[KDIM][DCH];   // cross-corr -> solution G in place
  __shared__ float2 sDinv;

  const int bf = blockIdx.x;
  const int b = bf / NF;
  const int tid = threadIdx.x;
  const size_t ybase = (size_t)bf * DCH * NTF;

  for (int idx = tid; idx < DCH * TFP; idx += 256) {
    int d = idx / TFP, t = idx % TFP;
    float2 v = make_float2(0.f, 0.f);
    if (t < NTF) v = Yg[ybase + (size_t)d * NTF + t];
    sY[d][t] = v;
  }
  const float eps = 1e-10f * __uint_as_float(maxb[b]);
  for (int t = tid; t < TFP; t += 256) {
    float v = 0.f;
    if (t < NTF) v = 1.0f / fmaxf(power[(size_t)bf * NTF + t], eps);
    sInvp[t] = v;
  }
  __syncthreads();

  // ---- Phase 1: 25 R tiles + 5 P tiles over 8 waves, f32 WMMA 16x16x4 ----
  const int wave = tid >> 5, lane = tid & 31;
  const int mrow = lane & 15, kb2 = (lane < 16) ? 0 : 2;
  for (int job = wave; job < 30; job += 8) {
    int tr, tc;
    bool isP;
    if (job < 25) { tr = job / 5; tc = job % 5; isP = false; }
    else          { tr = job - 25; tc = 0;       isP = true; }
    v8f accRe = {0, 0, 0, 0, 0, 0, 0, 0};
    v8f accIm = {0, 0, 0, 0, 0, 0, 0, 0};
    const int ka = tr * 16 + mrow;          // row of Yt (0..79)
    const int tauA = ka >> 3, dA = ka & 7;
    const int kbm = tc * 16 + mrow;         // col row for R-tiles
    const int tauB = kbm >> 3, dB = kbm & 7;
    const float mkP = (mrow < DCH) ? 1.f : 0.f;
    const int dnP = mrow & 7;
    for (int s = 0; s < 65; ++s) {
      const int t0 = 4 * s + kb2;
      // A = Yt_ip[ka][t0..t0+1] (invp folded, zero for t<0 history)
      int ts0 = t0 - DELAY - tauA;
      int ts1 = ts0 + 1;
      float m0 = (ts0 >= 0) ? sInvp[t0] : 0.f;
      float m1 = (ts1 >= 0) ? sInvp[t0 + 1] : 0.f;
      float2 y0 = sY[dA][ts0 < 0 ? 0 : ts0];
      float2 y1 = sY[dA][ts1 < 0 ? 0 : ts1];
      v2f Ar = {y0.x * m0, y1.x * m1};
      v2f Ai = {y0.y * m0, y1.y * m1};
      // B = Yt[kbm][t]  (R)   or   Y[dn][t] (P); conj folded into the math
      float2 b0, b1;
      if (!isP) {
        int u0 = t0 - DELAY - tauB;
        int u1 = u0 + 1;
        float n0 = (u0 >= 0) ? 1.f : 0.f;
        float n1 = (u1 >= 0) ? 1.f : 0.f;
        float2 z0 = sY[dB][u0 < 0 ? 0 : u0];
        float2 z1 = sY[dB][u1 < 0 ? 0 : u1];
        b0 = make_float2(z0.x * n0, z0.y * n0);
        b1 = make_float2(z1.x * n1, z1.y * n1);
      } else {
        float2 z0 = sY[dnP][t0], z1 = sY[dnP][t0 + 1];
        b0 = make_float2(z0.x * mkP, z0.y * mkP);
        b1 = make_float2(z1.x * mkP, z1.y * mkP);
      }
      v2f Byr = {b0.x, b1.x};
      v2f Byi = {b0.y, b1.y};
      v2f Arn = -Ar;  // manual negation (f32 WMMA NEG bits are C-only)
      // R[k,l] = sum_t A * conj(B): Re = Ar*Br + Ai*Bi ; Im = Ai*Br - Ar*Bi
      accRe = wmma4(Ar, Byr, accRe);
      accRe = wmma4(Ai, Byi, accRe);
      accIm = wmma4(Ai, Byr, accIm);
      accIm = wmma4(Arn, Byi, accIm);
    }
    const int col = mrow, rb = (lane < 16) ? 0 : 8;
#pragma unroll
    for (int i = 0; i < 8; ++i) {
      int row = tr * 16 + rb + i;
      if (!isP) sR[row][tc * 16 + col] = make_float2(accRe[i], accIm[i]);
      else if (col < DCH) sP[row][col] = make_float2(accRe[i], accIm[i]);
    }
  }
  __syncthreads();

  // ---- Phase 2: Gaussian elimination (complex, no pivoting) ----
  for (int j = 0; j < KDIM - 1; ++j) {
    if (tid == 0) {
      float2 dg = sR[j][j];
      float s2 = fmaxf(dg.x * dg.x + dg.y * dg.y, 1e-30f);
      sDinv = make_float2(dg.x / s2, -dg.y / s2);
    }
    __syncthreads();
    const int i = j + 1 + tid;
    if (i < KDIM) {
      float2 dv = sDinv;
      float2 rij = sR[i][j];
      float2 fct = make_float2(rij.x * dv.x - rij.y * dv.y,
                               rij.x * dv.y + rij.y * dv.x);
      for (int cc = j + 1; cc < KDIM; ++cc) {
        float2 u = sR[j][cc], w = sR[i][cc];
        w.x -= fct.x * u.x - fct.y * u.y;
        w.y -= fct.x * u.y + fct.y * u.x;
        sR[i][cc] = w;
      }
      for (int dd = 0; dd < DCH; ++dd) {
        float2 u = sP[j][dd], w = sP[i][dd];
        w.x -= fct.x * u.x - fct.y * u.y;
        w.y -= fct.x * u.y + fct.y * u.x;
        sP[i][dd] = w;
      }
    }
    __syncthreads();
  }
  // back substitution (column sweep); solution G overwrites sP
  for (int j = KDIM - 1; j >= 0; --j) {
    if (tid == 0) {
      float2 dg = sR[j][j];
      float s2 = fmaxf(dg.x * dg.x + dg.y * dg.y, 1e-30f);
      sDinv = make_float2(dg.x / s2, -dg.y / s2);
    }
    __syncthreads();
    if (tid < DCH) {
      float2 p = sP[j][tid], dv = sDinv;
      sP[j][tid] = make_float2(p.x * dv.x - p.y * dv.y,
                               p.x * dv.y + p.y * dv.x);
    }
    __syncthreads();
    for (int idx = tid; idx < j * DCH; idx += 256) {
      int i = idx >> 3, dd = idx & 7;
      float2 u = sR[i][j], g = sP[j][dd], w = sP[i][dd];
      w.x -= u.x * g.x - u.y * g.y;
      w.y -= u.x * g.y + u.y * g.x;
      sP[i][dd] = w;
    }
    __syncthreads();
  }

  // ---- Phase 3: X = Y - G^H Yt ----
  for (int idx = tid; idx < DCH * NTF; idx += 256) {
    int d = idx / NTF, t = idx % NTF;
    float2 acc = sY[d][t];
    for (int k = 0; k < KDIM; ++k) {
      int tau = k >> 3, dd = k & 7;
      int ts = t - DELAY - tau;
      if (ts >= 0) {
        float2 y = sY[dd][ts];
        float2 g = sP[k][d];  // conj(g) * y
        acc.x -= g.x * y.x + g.y * y.y;
        acc.y -= g.x * y.y - g.y * y.x;
      }
    }
    Xg[ybase + (size_t)d * NTF + t] = acc;
  }
}

// ---------------------------------------------------------------------------
// ISTFT frames: wframes(512 x Tf) = ICre*Zre - ICim*Zim via WMMA per (b,d).
__global__ __launch_bounds__(256) void k_ifft(const float2* __restrict__ Xg,
                                              const float* __restrict__ ICre,
                                              const float* __restrict__ ICim,
                                              float* __restrict__ Wf) {
  const int tid = threadIdx.x, wave = tid >> 5, lane = tid & 31;
  const int gj = blockIdx.x * 8 + wave;  // 32 * (32*17) jobs
  const int bd = gj / 544;
  const int tile = gj % 544;
  const int n0 = (tile / 17) * 16;
  const int tt0 = (tile % 17) * 16;
  const int b = bd >> 3, d = bd & 7;
  const int mrow = lane & 15, kb2 = (lane < 16) ? 0 : 2;

  v8f acc = {0, 0, 0, 0, 0, 0, 0, 0};
  const int t = tt0 + mrow;
  const int tc = (t < NTF) ? t : (NTF - 1);
  const float tm = (t < NTF) ? 1.f : 0.f;
  for (int s = 0; s < 65; ++s) {
    const int f0 = 4 * s + kb2;
    v2f Are = {ICre[(n0 + mrow) * TFP + f0], ICre[(n0 + mrow) * TFP + f0 + 1]};
    v2f Ain = {-ICim[(n0 + mrow) * TFP + f0], -ICim[(n0 + mrow) * TFP + f0 + 1]};
    int fa = (f0 < NF) ? f0 : (NF - 1);
    int fb = (f0 + 1 < NF) ? (f0 + 1) : (NF - 1);
    float ma = ((f0 < NF) ? 1.f : 0.f) * tm;
    float mb = ((f0 + 1 < NF) ? 1.f : 0.f) * tm;
    float2 z0 = Xg[((size_t)(b * NF + fa) * DCH + d) * NTF + tc];
    float2 z1 = Xg[((size_t)(b * NF + fb) * DCH + d) * NTF + tc];
    v2f Bre = {z0.x * ma, z1.x * mb};
    v2f Bim = {z0.y * ma, z1.y * mb};
    acc = wmma4(Are, Bre, acc);
    acc = wmma4(Ain, Bim, acc);
  }
  const int col = mrow, rb = (lane < 16) ? 0 : 8;
  const int tt = tt0 + col;
#pragma unroll
  for (int i = 0; i < 8; ++i) {
    int n = n0 + rb + i;
    if (tt < NTF) Wf[((size_t)bd * NTF + tt) * NFFT + n] = acc[i];
  }
}

// ---------------------------------------------------------------------------
// Deterministic OLA gather (<=4 frames per sample) + window^2 normalization.
__global__ __launch_bounds__(256) void k_ola(const float* __restrict__ Wf,
                                             const float* __restrict__ win,
                                             float* __restrict__ out) {
  __shared__ float sw[NFFT];
  for (int i = threadIdx.x; i < NFFT; i += 256) sw[i] = win[i];
  __syncthreads();
  const int gid = blockIdx.x * 256 + threadIdx.x;
  const int bd = gid >> 15;
  const int n = gid & (TLEN - 1);
  const int p = n + NFFT / 2;  // position in padded OLA signal
  int a = p - (NFFT - 1);
  int jmin = (a > 0) ? ((a + HOP - 1) >> 7) : 0;
  int jmax = p >> 7;
  if (jmax > NTF - 1) jmax = NTF - 1;
  float acc = 0.f, wsum = 0.f;
  for (int j = jmin; j <= jmax; ++j) {
    int off = p - j * HOP;
    acc += Wf[((size_t)bd * NTF + j) * NFFT + off];
    float w = sw[off];
    wsum += w * w;
  }
  out[gid] = acc / fmaxf(wsum, 1e-11f);
}

// ---------------------------------------------------------------------------
extern "C" void kernel_launch(void* const* d_in, const int* in_sizes, int n_in,
                              void* d_out, int out_size, void* d_ws,
                              size_t ws_size, hipStream_t stream) {
  const float* audio = (const float*)d_in[0];
  float* ws = (float*)d_ws;
  float* win = ws + OFF_WIN;
  float* Cre = ws + OFF_CRE;
  float* Cim = ws + OFF_CIM;
  float* ICre = ws + OFF_ICRE;
  float* ICim = ws + OFF_ICIM;
  float2* Yg = (float2*)(ws + OFF_Y);
  float2* Xg = (float2*)(ws + OFF_X);
  float* power = ws + OFF_POW;
  unsigned* maxb = (unsigned*)(ws + OFF_MAX);
  float* Wf = ws + OFF_WF;  // aliases Y (Y is dead before k_ifft)

  k_init<<<544, 256, 0, stream>>>(win, Cre, Cim, ICre, ICim);
  k_stft<<<1156, 256, 0, stream>>>(audio, win, Cre, Cim, Yg);
  for (int it = 0; it < ITERS; ++it) {
    k_zeromax<<<1, 32, 0, stream>>>(maxb);
    k_power<<<NB * NF, 256, 0, stream>>>(it == 0 ? Yg : Xg, power, maxb);
    k_wpe<<<NB * NF, 256, 0, stream>>>(Yg, power, maxb, Xg);
  }
  k_ifft<<<2176, 256, 0, stream>>>(Xg, ICre, ICim, Wf);
  k_ola<<<4096, 256, 0, stream>>>(Wf, win, (float*)d_out);
}